// project_C_grasp_batch_multi_90237262889318
// MI455X (gfx1250) — compile-verified
//
#include <hip/hip_runtime.h>
#include <math.h>

// Problem constants from the reference.
constexpr int BB = 64;
constexpr int NN = 100000;
constexpr int GG = 2048;
constexpr float A_STIFF = 100.0f;

typedef float f4 __attribute__((ext_vector_type(4)));

// ---------------------------------------------------------------------------
// Kernel 1: streaming copy V_predict -> V_new (first B*N*3 floats of d_out).
// Block-uniform bounds split: full blocks (4687 of 4688) run an unconditional
// clause of 4x global_load_b128 th:TH_LOAD_NT followed by 4x
// global_store_b128 th:TH_STORE_NT (2KB in flight per wave -> HBM saturation
// at modest residency); only the last block takes the guarded tail path.
// NT hints keep the 2x76.8MB stream out of WGP$/L2.
// ---------------------------------------------------------------------------
__global__ void vcopy_nt_kernel(const f4* __restrict__ src,
                                f4* __restrict__ dst,
                                int n4) {
    const int tpb        = blockDim.x;
    const int perBlk     = tpb * 4;
    const int blockStart = blockIdx.x * perBlk;
    const int base       = blockStart + threadIdx.x;

    const int i0 = base;
    const int i1 = base + tpb;
    const int i2 = base + 2 * tpb;
    const int i3 = base + 3 * tpb;

    if (blockStart + perBlk <= n4) {
        // Hot path: scalar-uniform, no per-lane predication -> loads clause.
        f4 v0 = __builtin_nontemporal_load(src + i0);
        f4 v1 = __builtin_nontemporal_load(src + i1);
        f4 v2 = __builtin_nontemporal_load(src + i2);
        f4 v3 = __builtin_nontemporal_load(src + i3);
        __builtin_nontemporal_store(v0, dst + i0);
        __builtin_nontemporal_store(v1, dst + i1);
        __builtin_nontemporal_store(v2, dst + i2);
        __builtin_nontemporal_store(v3, dst + i3);
    } else {
        // Tail block only.
        if (i0 < n4) { f4 v = __builtin_nontemporal_load(src + i0); __builtin_nontemporal_store(v, dst + i0); }
        if (i1 < n4) { f4 v = __builtin_nontemporal_load(src + i1); __builtin_nontemporal_store(v, dst + i1); }
        if (i2 < n4) { f4 v = __builtin_nontemporal_load(src + i2); __builtin_nontemporal_store(v, dst + i2); }
        if (i3 < n4) { f4 v = __builtin_nontemporal_load(src + i3); __builtin_nontemporal_store(v, dst + i3); }
    }
}

// ---------------------------------------------------------------------------
// Kernel 2: one lane per (b,g) constraint. Indices are unique per batch, so
// the scatter is a plain store that overwrites the copied row with
// V_predict + delta (exactly what the reference's .at[].add computes).
// Runs after kernel 1 on the same stream -> ordering guaranteed.
// ---------------------------------------------------------------------------
__global__ void grasp_project_kernel(const float* __restrict__ V_predict,
                                     const float* __restrict__ L,
                                     const float* __restrict__ grasp_points,
                                     const float* __restrict__ V_w,
                                     const float* __restrict__ C_grasp_d,
                                     const int*   __restrict__ C_grasp,
                                     float* __restrict__ V_out,
                                     float* __restrict__ L_out) {
    int t = blockIdx.x * blockDim.x + threadIdx.x;
    if (t >= BB * GG) return;

    const int b   = t / GG;          // batch
    const int idx = C_grasp[t];      // particle index within batch (unique per batch)

    const long vbase = ((long)b * NN + idx) * 3L;
    const long gbase = (long)t * 3L;

    // Gathered edge vector Nv = V_predict[b, idx] - grasp_points[b, g]
    const float vx = V_predict[vbase + 0];
    const float vy = V_predict[vbase + 1];
    const float vz = V_predict[vbase + 2];
    const float nx = vx - grasp_points[gbase + 0];
    const float ny = vy - grasp_points[gbase + 1];
    const float nz = vz - grasp_points[gbase + 2];

    const float D  = sqrtf(nx * nx + ny * ny + nz * nz);
    const float C  = D - C_grasp_d[t];

    const float Sg = V_w[(long)b * NN + idx];
    const float S  = (Sg == 0.0f) ? INFINITY : Sg;   // pinned -> zero correction

    const float Lv      = L[t];
    const float L_delta = (-C - A_STIFF * Lv) / (S + A_STIFF);
    L_out[t] = Lv + L_delta;

    // delta = Sg * L_delta * (Nv / D); written as corrected absolute position.
    const float scale = Sg * L_delta / D;
    V_out[vbase + 0] = vx + scale * nx;
    V_out[vbase + 1] = vy + scale * ny;
    V_out[vbase + 2] = vz + scale * nz;
}

extern "C" void kernel_launch(void* const* d_in, const int* in_sizes, int n_in,
                              void* d_out, int out_size, void* d_ws, size_t ws_size,
                              hipStream_t stream) {
    // setup_inputs() order:
    //   0: V_predict [B,N,3] f32
    //   1: L         [B,G,1] f32
    //   2: grasp_points [B,G,3] f32
    //   3: V_w       [B,N,1] f32
    //   4: C_grasp_d [B,G,1] f32
    //   5: C_grasp   [B,G]   int
    const float* V_predict    = (const float*)d_in[0];
    const float* L            = (const float*)d_in[1];
    const float* grasp_points = (const float*)d_in[2];
    const float* V_w          = (const float*)d_in[3];
    const float* C_grasp_d    = (const float*)d_in[4];
    const int*   C_grasp      = (const int*)  d_in[5];

    // d_out = [V_new (B*N*3 f32) | L_new (B*G f32)]
    float* V_out = (float*)d_out;
    float* L_out = V_out + (size_t)BB * NN * 3;

    // Streaming copy: 4.8M float4 elements, 4 per lane.
    const int n4 = (BB * NN * 3) / 4;
    {
        const int threads = 256;                  // 8 wave32 waves per block
        const int perBlk  = threads * 4;
        const int blocks  = (n4 + perBlk - 1) / perBlk;
        vcopy_nt_kernel<<<blocks, threads, 0, stream>>>(
            (const f4*)V_predict, (f4*)V_out, n4);
    }

    // Constraint projection: 131,072 lanes.
    {
        const int nc      = BB * GG;
        const int threads = 256;
        const int blocks  = (nc + threads - 1) / threads;
        grasp_project_kernel<<<blocks, threads, 0, stream>>>(
            V_predict, L, grasp_points, V_w, C_grasp_d, C_grasp, V_out, L_out);
    }
}